// GCN_45715631898727
// MI455X (gfx1250) — compile-verified
//
#include <hip/hip_runtime.h>
#include <hip/hip_bf16.h>

typedef __attribute__((ext_vector_type(2))) float v2f;
typedef __attribute__((ext_vector_type(8))) float v8f;

#define NGRAPHS 256
#define HPAD    64    // padded feature stride for hidden layers
#define HOUT    50    // real hidden width
#define FIN     136   // input feature width

// ---------------------------------------------------------------- utilities

__global__ void k_init(float* __restrict__ deg, float* __restrict__ pooled,
                       int* __restrict__ cnt, int N) {
  int i = blockIdx.x * blockDim.x + threadIdx.x;
  if (i < N) deg[i] = 1.0f;                 // self-loop contributes 1 to degree
  if (i < NGRAPHS * HPAD) pooled[i] = 0.0f;
  if (i < NGRAPHS) cnt[i] = 0;
}

__global__ void k_deg(const int* __restrict__ dst, float* __restrict__ deg, int E) {
  int e = blockIdx.x * blockDim.x + threadIdx.x;
  if (e < E) atomicAdd(&deg[dst[e]], 1.0f);
}

__global__ void k_dinv(float* __restrict__ deg, int N) {
  int i = blockIdx.x * blockDim.x + threadIdx.x;
  if (i < N) deg[i] = rsqrtf(deg[i]);       // deg >= 1 always (self-loop)
}

// pack W [K x Ncols] row-major into Wp [Kp x 64] with zero fill
__global__ void k_packW(const float* __restrict__ W, float* __restrict__ Wp,
                        int K, int Ncols, int Kp) {
  int i = blockIdx.x * blockDim.x + threadIdx.x;
  if (i >= Kp * HPAD) return;
  int k = i >> 6, n = i & 63;
  Wp[i] = (k < K && n < Ncols) ? W[k * Ncols + n] : 0.0f;
}

// ---------------------------------------------------------------- WMMA GEMM
// Y[M x 64] = X[M x ldx] * W[K x 64], one wave per 16-row tile,
// V_WMMA_F32_16X16X4_F32, K must be a multiple of 4.
__global__ void k_gemm_wmma(const float* __restrict__ X, const float* __restrict__ W,
                            float* __restrict__ Y, int M, int K, int ldx) {
  int wave = threadIdx.x >> 5;
  int lane = threadIdx.x & 31;
  int tile = blockIdx.x * (blockDim.x >> 5) + wave;
  int m_tiles = (M + 15) >> 4;
  if (tile >= m_tiles) return;              // whole-wave uniform exit; EXEC stays full

  int row0 = tile << 4;
  int half = lane >> 4;                     // 0: lanes 0-15, 1: lanes 16-31
  int l15  = lane & 15;
  int arow = row0 + l15;
  if (arow >= M) arow = M - 1;              // clamp (dup read; stores are guarded)
  const float* xrow = X + (size_t)arow * ldx;

  v8f c0 = {}, c1 = {}, c2 = {}, c3 = {};
  for (int k = 0; k < K; k += 4) {
    int ka = k + 2 * half;                  // A 16x4 layout: lane-half selects K pair
    v2f a;
    a.x = xrow[ka];
    a.y = xrow[ka + 1];
    const float* w0 = W + (size_t)ka * HPAD + l15;
    const float* w1 = w0 + HPAD;
    v2f b0 = { w0[0],  w1[0]  };
    v2f b1 = { w0[16], w1[16] };
    v2f b2 = { w0[32], w1[32] };
    v2f b3 = { w0[48], w1[48] };
    c0 = __builtin_amdgcn_wmma_f32_16x16x4_f32(false, a, false, b0, (short)0, c0, false, false);
    c1 = __builtin_amdgcn_wmma_f32_16x16x4_f32(false, a, false, b1, (short)0, c1, false, false);
    c2 = __builtin_amdgcn_wmma_f32_16x16x4_f32(false, a, false, b2, (short)0, c2, false, false);
    c3 = __builtin_amdgcn_wmma_f32_16x16x4_f32(false, a, false, b3, (short)0, c3, false, false);
  }

  // C/D layout: VGPR r -> row (8*half + r), lane&15 -> column; coalesced stores
  float* yb = Y + (size_t)(row0 + 8 * half) * HPAD + l15;
#pragma unroll
  for (int r = 0; r < 8; ++r) {
    if (row0 + 8 * half + r < M) {
      yb[(size_t)r * HPAD + 0]  = c0[r];
      yb[(size_t)r * HPAD + 16] = c1[r];
      yb[(size_t)r * HPAD + 32] = c2[r];
      yb[(size_t)r * HPAD + 48] = c3[r];
    }
  }
}

// ---------------------------------------------------------------- aggregation

// acc[v] = hg[v] * dinv[v]   (self-loop term; later scaled by dinv[v] again)
__global__ void k_selfinit(const float* __restrict__ hg, const float* __restrict__ dinv,
                           float* __restrict__ acc, int N) {
  long long i = (long long)blockIdx.x * blockDim.x + threadIdx.x;
  if (i >= (long long)N * HPAD) return;
  int v = (int)(i >> 6);
  acc[i] = hg[i] * dinv[v];
}

// per (edge, feature): acc[dst] += hg[src] * dinv[src]
__global__ void k_edge_agg(const int* __restrict__ src, const int* __restrict__ dst,
                           const float* __restrict__ hg, const float* __restrict__ dinv,
                           float* __restrict__ acc, int E) {
  long long idx = (long long)blockIdx.x * blockDim.x + threadIdx.x;
  long long e = idx >> 6;
  int f = (int)(idx & 63);
  if (e >= E || f >= HOUT) return;
  int s = src[e], d = dst[e];
  float v = hg[(size_t)s * HPAD + f] * dinv[s];
  atomicAdd(&acc[(size_t)d * HPAD + f], v);
}

// h[v] = relu(acc[v] * dinv[v] + bias)  (in place, zero the pad columns)
__global__ void k_epilogue(float* __restrict__ acc, const float* __restrict__ dinv,
                           const float* __restrict__ bias, int N) {
  long long i = (long long)blockIdx.x * blockDim.x + threadIdx.x;
  if (i >= (long long)N * HPAD) return;
  int v = (int)(i >> 6);
  int f = (int)(i & 63);
  float t = 0.0f;
  if (f < HOUT) t = fmaxf(acc[i] * dinv[v] + bias[f], 0.0f);
  acc[i] = t;
}

// ---------------------------------------------------------------- pooling + head

__global__ void k_pool(const float* __restrict__ h, const int* __restrict__ batch,
                       float* __restrict__ pooled, int* __restrict__ cnt, int N) {
  long long i = (long long)blockIdx.x * blockDim.x + threadIdx.x;
  if (i >= (long long)N * HPAD) return;
  int n = (int)(i >> 6);
  int f = (int)(i & 63);
  int g = batch[n];
  if (f < HOUT) atomicAdd(&pooled[(size_t)g * HPAD + f], h[i]);
  if (f == 0)   atomicAdd(&cnt[g], 1);
}

__global__ void k_head(const float* __restrict__ pooled, const int* __restrict__ cnt,
                       const float* __restrict__ Wl, const float* __restrict__ bl,
                       float* __restrict__ out) {
  int g = blockIdx.x * blockDim.x + threadIdx.x;
  if (g >= NGRAPHS) return;
  float c = fmaxf((float)cnt[g], 1.0f);
  float s = 0.0f;
  for (int f = 0; f < HOUT; ++f)
    s += (pooled[(size_t)g * HPAD + f] / c) * Wl[f];
  float z = s + bl[0];
  out[g] = 1.0f / (1.0f + expf(-z));
}

// ---------------------------------------------------------------- launcher

extern "C" void kernel_launch(void* const* d_in, const int* in_sizes, int n_in,
                              void* d_out, int out_size, void* d_ws, size_t ws_size,
                              hipStream_t stream) {
  const float* x     = (const float*)d_in[0];
  const int*   edge  = (const int*)d_in[1];
  const int*   batch = (const int*)d_in[2];
  const float* W1    = (const float*)d_in[3];
  const float* b1    = (const float*)d_in[4];
  const float* W2    = (const float*)d_in[5];
  const float* b2    = (const float*)d_in[6];
  const float* Wl    = (const float*)d_in[7];
  const float* bl    = (const float*)d_in[8];
  float*       out   = (float*)d_out;

  const int N = in_sizes[0] / FIN;
  const int E = in_sizes[1] / 2;
  const int* esrc = edge;
  const int* edst = edge + E;

  // workspace carve-up
  char* ws = (char*)d_ws;
  auto alignup = [](size_t v) { return (v + 255) & ~(size_t)255; };
  size_t o = 0;
  float* A      = (float*)(ws + o); o = alignup(o + (size_t)N * HPAD * 4);  // GEMM output
  float* B      = (float*)(ws + o); o = alignup(o + (size_t)N * HPAD * 4);  // agg acc / activations
  float* dinv   = (float*)(ws + o); o = alignup(o + (size_t)N * 4);
  float* W1p    = (float*)(ws + o); o = alignup(o + (size_t)FIN * HPAD * 4);
  float* W2p    = (float*)(ws + o); o = alignup(o + (size_t)HPAD * HPAD * 4);
  float* pooled = (float*)(ws + o); o = alignup(o + (size_t)NGRAPHS * HPAD * 4);
  int*   cnt    = (int*)  (ws + o); o = alignup(o + (size_t)NGRAPHS * 4);
  (void)ws_size; (void)n_in; (void)out_size;

  const int T = 256;
  const int gN   = (N + T - 1) / T;
  const int gE   = (E + T - 1) / T;
  const int gNH  = (int)(((long long)N * HPAD + T - 1) / T);
  const int gEH  = (int)(((long long)E * HPAD + T - 1) / T);
  const int m_tiles = (N + 15) / 16;
  const int gGemm   = (m_tiles + 3) / 4;     // 4 waves (128 threads) per block

  // degrees + norms
  k_init <<<gN, T, 0, stream>>>(dinv, pooled, cnt, N);
  k_deg  <<<gE, T, 0, stream>>>(edst, dinv, E);
  k_dinv <<<gN, T, 0, stream>>>(dinv, N);

  // pack weights
  k_packW<<<(FIN * HPAD + T - 1) / T, T, 0, stream>>>(W1, W1p, FIN, HOUT, FIN);
  k_packW<<<(HPAD * HPAD + T - 1) / T, T, 0, stream>>>(W2, W2p, HOUT, HOUT, HPAD);

  // ---- layer 1: h1 = relu(norm-agg(x @ W1) + b1)
  k_gemm_wmma<<<gGemm, 128, 0, stream>>>(x, W1p, A, N, FIN, FIN);
  k_selfinit <<<gNH, T, 0, stream>>>(A, dinv, B, N);
  k_edge_agg <<<gEH, T, 0, stream>>>(esrc, edst, A, dinv, B, E);
  k_epilogue <<<gNH, T, 0, stream>>>(B, dinv, b1, N);

  // ---- layer 2: h2 = relu(norm-agg(h1 @ W2) + b2)   (B consumed by GEMM, then reused)
  k_gemm_wmma<<<gGemm, 128, 0, stream>>>(B, W2p, A, N, HPAD, HPAD);
  k_selfinit <<<gNH, T, 0, stream>>>(A, dinv, B, N);
  k_edge_agg <<<gEH, T, 0, stream>>>(esrc, edst, A, dinv, B, E);
  k_epilogue <<<gNH, T, 0, stream>>>(B, dinv, b2, N);

  // ---- pool + head
  k_pool<<<gNH, T, 0, stream>>>(B, batch, pooled, cnt, N);
  k_head<<<1, NGRAPHS, 0, stream>>>(pooled, cnt, Wl, bl, out);
}